// TensorBoard_4423816315110
// MI455X (gfx1250) — compile-verified
//
#include <hip/hip_runtime.h>
#include <hip/hip_bf16.h>
#include <stdint.h>

// Problem constants (from reference setup: B=16384, BS=19, G=8, S=4)
#define BS2     361          // 19*19 positions per board == history length
#define GMAX    8            // groups per board
#define TBL     1024         // LDS hash table slots (power of 2, > 361)
#define TBL_MASK (TBL - 1)
#define BLOCK_T 384          // 12 wave32 waves per board

// ---- CDNA5 feature detection (device pass only) ----
#if defined(__gfx1250__) && __has_builtin(__builtin_amdgcn_global_load_async_to_lds_b32) && __has_builtin(__builtin_amdgcn_s_wait_asynccnt)
#define HAVE_ASYNC_LDS 1
#else
#define HAVE_ASYNC_LDS 0
#endif

#if defined(__gfx1250__) && __has_builtin(__builtin_amdgcn_tensor_load_to_lds) && __has_builtin(__builtin_amdgcn_s_wait_tensorcnt)
#define HAVE_TDM 1
#else
#define HAVE_TDM 0
#endif

typedef __attribute__((address_space(1))) int as1_int;   // global
typedef __attribute__((address_space(3))) int as3_int;   // LDS
typedef uint32_t v4u __attribute__((ext_vector_type(4)));
typedef int32_t  v8i __attribute__((ext_vector_type(8)));
typedef int32_t  v4i __attribute__((ext_vector_type(4)));

__device__ __forceinline__ uint32_t fibhash(uint32_t v) {
    // Fibonacci hashing -> top 10 bits -> [0, 1024)
    return (v * 2654435769u) >> 22;
}

__global__ __launch_bounds__(BLOCK_T) void go_repeat_mask_kernel(
    const float* __restrict__ logits,
    const unsigned char* __restrict__ legal,      // numpy bool -> 1 byte
    const int* __restrict__ cur_player,
    const int* __restrict__ cur_hash,
    const int* __restrict__ hash_hist,
    const int* __restrict__ move_count,
    const int* __restrict__ ZposT,                // (3, N2)
    const int* __restrict__ sgi,                  // stone_global_index (K)
    const int* __restrict__ sgp,                  // stone_global_pointer (R+1)
    const int* __restrict__ ggp,                  // group_global_pointer_per_board (B+1)
    const int* __restrict__ cgli,                 // captured_group_local_index (B, N2, 4)
    float* __restrict__ out,
    int n2)
{
    __shared__ int hist[BS2];                       // board's hash history row
    __shared__ int table[TBL];                      // hash table: 0 = empty, else j+1
    __shared__ int gxor[GMAX];                      // per-group capture XOR
    __shared__ __align__(16) int cap_lds[BS2 * 4];  // TDM-staged captured indices

    const int b = blockIdx.x;
    const int tid = threadIdx.x;
    const size_t rowbase = (size_t)b * BS2;

    // ---------------- Kick off async data movement first ----------------
#if HAVE_TDM
    if (tid == 0) {
        // D# for a 1D tile: 1444 int32 of this board's captured_group_local_index.
        const uint32_t nelem = (uint32_t)(BS2 * 4);            // 1444 elements (fits 16b tile_dim0)
        const uint64_t ga = (uint64_t)(uintptr_t)(cgli + rowbase * 4);
        const uint32_t lds_off = (uint32_t)(uintptr_t)(&cap_lds[0]);
        v4u g0; v8i g1; v4i g2 = {0, 0, 0, 0}; v4i g3 = {0, 0, 0, 0};
        g0[0] = 1u;                                            // count=1, is_restore=0, gather=0
        g0[1] = lds_off;                                       // lds_addr (bytes)
        g0[2] = (uint32_t)ga;                                  // global_addr[31:0]
        g0[3] = ((uint32_t)(ga >> 32) & 0x01FFFFFFu) | (2u << 30); // addr[56:32], type=2 (image)
        g1[0] = (int)(2u << 16);                               // wg_mask=0, data_size=2 (4B)
        g1[1] = (int)((nelem & 0xFFFFu) << 16);                // tensor_dim0[15:0]
        g1[2] = (int)(((nelem >> 16) & 0xFFFFu) | (1u << 16)); // tensor_dim0[31:16], tensor_dim1=1
        g1[3] = (int)((nelem & 0xFFFFu) << 16);                // tile_dim0 = 1444
        g1[4] = 1;                                             // tile_dim1 = 1
        g1[5] = (int)nelem;                                    // tensor_dim0_stride lo32
        g1[6] = 0;
        g1[7] = 0;
#if __has_include(<hip/amd_detail/amd_gfx1250_TDM.h>)
        v8i g4 = {0, 0, 0, 0, 0, 0, 0, 0};                     // therock 6-arg form
        __builtin_amdgcn_tensor_load_to_lds(g0, g1, g2, g3, g4, 0);
#else
        __builtin_amdgcn_tensor_load_to_lds(g0, g1, g2, g3, 0); // ROCm 7.2 5-arg form
#endif
    }
#endif

#if HAVE_ASYNC_LDS
    if (tid < BS2) {
        __builtin_amdgcn_global_load_async_to_lds_b32(
            (as1_int*)(uintptr_t)(hash_hist + rowbase + tid),
            (as3_int*)(uint32_t)(uintptr_t)(&hist[tid]),
            0, 0);
    }
#else
    if (tid < BS2) hist[tid] = hash_hist[rowbase + tid];
#endif

    // Prefetch the tail-consumed rows while we build the hash table.
#if defined(__gfx1250__)
    if (tid < BS2) {
        __builtin_prefetch(logits + rowbase + tid, 0, 0);
        __builtin_prefetch(legal + rowbase + tid, 0, 0);
    }
#endif

    // ---------------- Overlapped VALU/LDS work ----------------
    for (int i = tid; i < TBL; i += BLOCK_T) table[i] = 0;
    if (tid < GMAX) gxor[tid] = 0;

    const int cp  = cur_player[b];
    const int gp0 = ggp[b];
    const int ng  = ggp[b + 1] - gp0;
    if (tid < GMAX && tid < ng) {
        const int g  = gp0 + tid;
        const int s0 = sgp[g], s1 = sgp[g + 1];
        const int opp_row = 2 - cp;              // ZposT row of opponent color
        int x = 0;
        for (int s = s0; s < s1; ++s) {
            const int si = sgi[s];
            x ^= ZposT[opp_row * n2 + si] ^ ZposT[si];
        }
        gxor[tid] = x;
    }

    int mc = move_count[b];
    mc = mc < 0 ? 0 : (mc > BS2 ? BS2 : mc);

    // ---------------- Wait for data movers, then build table ----------------
#if HAVE_ASYNC_LDS
    __builtin_amdgcn_s_wait_asynccnt(0);
#endif
#if HAVE_TDM
    __builtin_amdgcn_s_wait_tensorcnt(0);
#endif
    __syncthreads();

    // Insert valid history entries: slot holds (index+1); duplicates simply chain.
    for (int j = tid; j < mc; j += BLOCK_T) {
        uint32_t h = fibhash((uint32_t)hist[j]);
        const int key = j + 1;
        for (;;) {
            const int prev = atomicCAS(&table[h], 0, key);   // ds_cmpstore_rtn_b32
            if (prev == 0) break;
            h = (h + 1) & TBL_MASK;
        }
    }
    __syncthreads();

    // ---------------- Per-move candidate hash + membership + mask ----------------
    if (tid < BS2) {
        const int p = tid;
        const int pd = ZposT[p] ^ ZposT[(cp == 0 ? 1 : 2) * n2 + p];

#if HAVE_TDM
        const int4 cv = ((const int4*)cap_lds)[p];           // ds_load_b128
#else
        const int4 cv = ((const int4*)cgli)[rowbase + p];    // global_load_b128
#endif
        int cd = 0;
        if (cv.x >= 0) cd ^= gxor[cv.x];
        if (cv.y >= 0) cd ^= gxor[cv.y];
        if (cv.z >= 0) cd ^= gxor[cv.z];
        if (cv.w >= 0) cd ^= gxor[cv.w];

        const int cand = cur_hash[b] ^ pd ^ cd;

        bool repeat = false;
        uint32_t h = fibhash((uint32_t)cand);
        for (;;) {
            const int s = table[h];
            if (s == 0) break;
            if (hist[s - 1] == cand) { repeat = true; break; }
            h = (h + 1) & TBL_MASK;
        }
        // Reference edge case: padding entries are I32MIN in the sorted history,
        // so candidate == I32MIN matches whenever any padding exists.
        if (cand == (int)0x80000000 && mc < BS2) repeat = true;

        const bool ok = (legal[rowbase + p] != 0) && !repeat;
        out[rowbase + p] = ok ? logits[rowbase + p] : -1000000000.0f;
    }
}

extern "C" void kernel_launch(void* const* d_in, const int* in_sizes, int n_in,
                              void* d_out, int out_size, void* d_ws, size_t ws_size,
                              hipStream_t stream) {
    const float*         logits     = (const float*)d_in[0];
    const unsigned char* legal      = (const unsigned char*)d_in[1];
    const int*           cur_player = (const int*)d_in[2];
    const int*           cur_hash   = (const int*)d_in[3];
    const int*           hash_hist  = (const int*)d_in[4];
    const int*           move_count = (const int*)d_in[5];
    const int*           ZposT      = (const int*)d_in[6];
    const int*           sgi        = (const int*)d_in[7];
    const int*           sgp        = (const int*)d_in[8];
    const int*           ggp        = (const int*)d_in[9];
    const int*           cgli       = (const int*)d_in[10];
    float*               out        = (float*)d_out;

    const int B  = in_sizes[2];          // boards
    const int n2 = in_sizes[6] / 3;      // 361

    go_repeat_mask_kernel<<<B, BLOCK_T, 0, stream>>>(
        logits, legal, cur_player, cur_hash, hash_hist, move_count,
        ZposT, sgi, sgp, ggp, cgli, out, n2);
}